// Llama4TextDecoderLayer_12627203850421
// MI455X (gfx1250) — compile-verified
//
#include <hip/hip_runtime.h>
#include <hip/hip_bf16.h>

// Problem sizes (fixed by the reference).
constexpr int kT = 2048;   // tokens
constexpr int kH = 2048;   // hidden
constexpr int kI = 4096;   // intermediate
constexpr int kE = 8;      // experts
constexpr float kEps = 1e-5f;

// GEMM tiling: 256 threads = 8 waves; per WG: M=64, N=128, K-chunk=32.
constexpr int BM = 64;
constexpr int BN = 128;
constexpr int BK = 32;
constexpr int AS = BK + 1;   // LDS stride for A rows
constexpr int BS = BN + 4;   // LDS stride for B rows (keeps float4 alignment)

typedef float v2f __attribute__((ext_vector_type(2)));
typedef float v8f __attribute__((ext_vector_type(8)));

// D = A(16x4) * B(4x16) + C, fp32 WMMA. 8-arg pattern per probe:
// (neg_a, A, neg_b, B, c_mod, C, reuse_a, reuse_b)
static __device__ __forceinline__ v8f wmma4(v2f a, v2f b, v8f c) {
  return __builtin_amdgcn_wmma_f32_16x16x4_f32(false, a, false, b, (short)0, c,
                                               false, false);
}

// ---------------------------------------------------------------------------
// Kernel 1: RMSNorm + router logits + top-1 (argmax) + sigmoid score.
// One 256-thread block per token.
// ---------------------------------------------------------------------------
__global__ __launch_bounds__(256) void rmsnorm_router_kernel(
    const float* __restrict__ x, const float* __restrict__ norm_w,
    const float* __restrict__ router_w, float* __restrict__ hbuf,
    float* __restrict__ score, int* __restrict__ eid) {
  const int t = blockIdx.x;
  const int tid = threadIdx.x;
  __shared__ float red[256];
  __shared__ float lg[kE];

  const float* xr = x + (size_t)t * kH;
  float xv[8];
  float ss = 0.f;
#pragma unroll
  for (int j = 0; j < 8; ++j) {
    xv[j] = xr[tid + j * 256];
    ss += xv[j] * xv[j];
  }
  red[tid] = ss;
  __syncthreads();
  for (int s = 128; s > 0; s >>= 1) {
    if (tid < s) red[tid] += red[tid + s];
    __syncthreads();
  }
  const float inv = rsqrtf(red[0] / (float)kH + kEps);
  __syncthreads();

  float part[kE];
#pragma unroll
  for (int e = 0; e < kE; ++e) part[e] = 0.f;
#pragma unroll
  for (int j = 0; j < 8; ++j) {
    const int idx = tid + j * 256;
    const float hv = xv[j] * inv * norm_w[idx];
    hbuf[(size_t)t * kH + idx] = hv;
#pragma unroll
    for (int e = 0; e < kE; ++e) part[e] += hv * router_w[(size_t)e * kH + idx];
  }
  for (int e = 0; e < kE; ++e) {
    red[tid] = part[e];
    __syncthreads();
    for (int s = 128; s > 0; s >>= 1) {
      if (tid < s) red[tid] += red[tid + s];
      __syncthreads();
    }
    if (tid == 0) lg[e] = red[0];
    __syncthreads();
  }
  if (tid == 0) {
    int best = 0;
    float bv = lg[0];
    for (int e = 1; e < kE; ++e)
      if (lg[e] > bv) { bv = lg[e]; best = e; }  // ties -> lowest idx (top_k)
    eid[t] = best;
    score[t] = 1.f / (1.f + expf(-bv));  // sigmoid of kept logit
  }
}

// ---------------------------------------------------------------------------
// Kernel 2: bucket tokens by expert (single workgroup, LDS atomics).
// List order within an expert is irrelevant to the final output.
// ---------------------------------------------------------------------------
__global__ __launch_bounds__(256) void bucket_kernel(
    const int* __restrict__ eid, int* __restrict__ tlist,
    int* __restrict__ ecnt, int* __restrict__ eoff) {
  __shared__ int cnt[kE], cur[kE], off[kE];
  const int tid = threadIdx.x;
  if (tid < kE) cnt[tid] = 0;
  __syncthreads();
  for (int t = tid; t < kT; t += 256) atomicAdd(&cnt[eid[t]], 1);
  __syncthreads();
  if (tid == 0) {
    int acc = 0;
    for (int e = 0; e < kE; ++e) { off[e] = acc; cur[e] = acc; acc += cnt[e]; }
  }
  __syncthreads();
  for (int t = tid; t < kT; t += 256) {
    const int pos = atomicAdd(&cur[eid[t]], 1);
    tlist[pos] = t;
  }
  if (tid < kE) { ecnt[tid] = cnt[tid]; eoff[tid] = off[tid]; }
}

// ---------------------------------------------------------------------------
// Kernel 3: gate_up GEMM + SiLU(gate)*up -> act[T, I].
// mode 0: shared expert (A = h rows, W [I,H] N-major/K-contiguous)
// mode 1: routed (A = h rows gathered via tlist, scaled by score;
//                 W = gate_up_proj[e] [H,2I] K-major/N-contiguous)
// ---------------------------------------------------------------------------
__global__ __launch_bounds__(256) void gateup_kernel(
    const float* __restrict__ hbuf, const float* __restrict__ score,
    const int* __restrict__ tlist, const int* __restrict__ ecnt,
    const int* __restrict__ eoff, const float* __restrict__ wg,
    const float* __restrict__ wu, const float* __restrict__ gup,
    float* __restrict__ act, int mode) {
  const int nb = blockIdx.x, mb = blockIdx.y, e = blockIdx.z;
  const int tid = threadIdx.x;
  int cnt = kT, base = 0;
  if (mode == 1) {
    cnt = ecnt[e];
    base = eoff[e];
    if (mb * BM >= cnt) return;
  }
  __shared__ float As[BM * AS];
  __shared__ float Bg[BK * BS];
  __shared__ float Bu[BK * BS];
  __shared__ int rowTok[BM];
  __shared__ float rowScale[BM];

  if (tid < BM) {
    const int m = mb * BM + tid;
    int tok; float sc;
    if (mode == 0) { tok = m; sc = 1.f; }
    else if (m < cnt) { tok = tlist[base + m]; sc = score[tok]; }
    else { tok = tlist[base]; sc = 0.f; }  // padded row contributes zeros
    rowTok[tid] = tok;
    rowScale[tid] = sc;
  }
  __syncthreads();

  const int lane = tid & 31, wave = tid >> 5;
  const int lrow = lane & 15;
  const int khalf = (lane >> 4) << 1;  // lanes 16-31 hold K+2,K+3
  const int cw = wave * 16 + lrow;     // this lane's column within BN

  v8f accG[4] = {};
  v8f accU[4] = {};

  const int ar = tid >> 2;            // A-fill: 4 threads/row, 8 floats each
  const int ac = (tid & 3) * 8;
  const size_t arow = (size_t)rowTok[ar] * kH;
  const float asc = rowScale[ar];

  for (int kb = 0; kb < kH; kb += BK) {
    {  // A tile (scaled)
      const float* src = hbuf + arow + kb + ac;
      const float4 v0 = *(const float4*)src;
      const float4 v1 = *(const float4*)(src + 4);
      float* dst = &As[ar * AS + ac];
      dst[0] = v0.x * asc; dst[1] = v0.y * asc; dst[2] = v0.z * asc; dst[3] = v0.w * asc;
      dst[4] = v1.x * asc; dst[5] = v1.y * asc; dst[6] = v1.z * asc; dst[7] = v1.w * asc;
    }
    if (mode == 0) {  // W rows are output columns; transpose into LDS
      const int n = tid >> 1, c0 = (tid & 1) * 16;
      const float* sg = wg + (size_t)(nb * BN + n) * kH + kb + c0;
      const float* su = wu + (size_t)(nb * BN + n) * kH + kb + c0;
#pragma unroll
      for (int q = 0; q < 4; ++q) {
        const float4 vg = *(const float4*)(sg + q * 4);
        const float4 vu = *(const float4*)(su + q * 4);
        const int kk = c0 + q * 4;
        Bg[(kk + 0) * BS + n] = vg.x; Bg[(kk + 1) * BS + n] = vg.y;
        Bg[(kk + 2) * BS + n] = vg.z; Bg[(kk + 3) * BS + n] = vg.w;
        Bu[(kk + 0) * BS + n] = vu.x; Bu[(kk + 1) * BS + n] = vu.y;
        Bu[(kk + 2) * BS + n] = vu.z; Bu[(kk + 3) * BS + n] = vu.w;
      }
    } else {  // gate_up_proj[e]: row k contiguous in n; gate at n, up at I+n
      const int kk = tid >> 3, c0 = (tid & 7) * 16;
      const float* row =
          gup + ((size_t)e * kH + kb + kk) * (size_t)(2 * kI) + nb * BN + c0;
#pragma unroll
      for (int q = 0; q < 4; ++q) {
        *(float4*)&Bg[kk * BS + c0 + q * 4] = *(const float4*)(row + q * 4);
        *(float4*)&Bu[kk * BS + c0 + q * 4] = *(const float4*)(row + kI + q * 4);
      }
    }
    __syncthreads();
#pragma unroll
    for (int k4 = 0; k4 < BK; k4 += 4) {
      const int k0 = k4 + khalf;
      v2f bg, bu;
      bg.x = Bg[k0 * BS + cw];       bg.y = Bg[(k0 + 1) * BS + cw];
      bu.x = Bu[k0 * BS + cw];       bu.y = Bu[(k0 + 1) * BS + cw];
#pragma unroll
      for (int ms = 0; ms < 4; ++ms) {
        v2f a;
        a.x = As[(ms * 16 + lrow) * AS + k0];
        a.y = As[(ms * 16 + lrow) * AS + k0 + 1];
        accG[ms] = wmma4(a, bg, accG[ms]);
        accU[ms] = wmma4(a, bu, accU[ms]);
      }
    }
    __syncthreads();
  }

  const int nglob = nb * BN + cw;
  const int mup = (lane >> 4) << 3;  // D: lanes 16-31 hold rows M+8
#pragma unroll
  for (int ms = 0; ms < 4; ++ms) {
#pragma unroll
    for (int i = 0; i < 8; ++i) {
      const int mloc = ms * 16 + i + mup;
      if (mode == 0 || (mb * BM + mloc) < cnt) {
        const float g = accG[ms][i];
        const float u = accU[ms][i];
        const float s = g / (1.f + expf(-g));  // silu(g)
        act[(size_t)rowTok[mloc] * kI + nglob] = u * s;
      }
    }
  }
}

// ---------------------------------------------------------------------------
// Kernel 4: down GEMM.
// mode 0: out = x + act @ shared_down_w^T   (W [H,I] N-major/K-contiguous)
// mode 1: out += act(gathered) @ down_proj[e] (W [I,H] K-major/N-contiguous)
// ---------------------------------------------------------------------------
__global__ __launch_bounds__(256) void down_kernel(
    const float* __restrict__ actb, const int* __restrict__ tlist,
    const int* __restrict__ ecnt, const int* __restrict__ eoff,
    const float* __restrict__ wd, const float* __restrict__ dp,
    const float* __restrict__ resid, float* __restrict__ outp, int mode) {
  const int nb = blockIdx.x, mb = blockIdx.y, e = blockIdx.z;
  const int tid = threadIdx.x;
  int cnt = kT, base = 0;
  if (mode == 1) {
    cnt = ecnt[e];
    base = eoff[e];
    if (mb * BM >= cnt) return;
  }
  __shared__ float As[BM * AS];
  __shared__ float Bs[BK * BS];
  __shared__ int rowTok[BM];

  if (tid < BM) {
    const int m = mb * BM + tid;
    rowTok[tid] = (mode == 0) ? m
                : (m < cnt ? tlist[base + m] : tlist[base]);
  }
  __syncthreads();

  const int lane = tid & 31, wave = tid >> 5;
  const int lrow = lane & 15;
  const int khalf = (lane >> 4) << 1;
  const int cw = wave * 16 + lrow;

  v8f acc[4] = {};

  const int ar = tid >> 2;
  const int ac = (tid & 3) * 8;
  const size_t arow = (size_t)rowTok[ar] * kI;

  for (int kb = 0; kb < kI; kb += BK) {
    {
      const float* src = actb + arow + kb + ac;
      const float4 v0 = *(const float4*)src;
      const float4 v1 = *(const float4*)(src + 4);
      float* dst = &As[ar * AS + ac];
      dst[0] = v0.x; dst[1] = v0.y; dst[2] = v0.z; dst[3] = v0.w;
      dst[4] = v1.x; dst[5] = v1.y; dst[6] = v1.z; dst[7] = v1.w;
    }
    if (mode == 0) {
      const int n = tid >> 1, c0 = (tid & 1) * 16;
      const float* sw = wd + (size_t)(nb * BN + n) * kI + kb + c0;
#pragma unroll
      for (int q = 0; q < 4; ++q) {
        const float4 v = *(const float4*)(sw + q * 4);
        const int kk = c0 + q * 4;
        Bs[(kk + 0) * BS + n] = v.x; Bs[(kk + 1) * BS + n] = v.y;
        Bs[(kk + 2) * BS + n] = v.z; Bs[(kk + 3) * BS + n] = v.w;
      }
    } else {
      const int kk = tid >> 3, c0 = (tid & 7) * 16;
      const float* row =
          dp + ((size_t)e * kI + kb + kk) * (size_t)kH + nb * BN + c0;
#pragma unroll
      for (int q = 0; q < 4; ++q)
        *(float4*)&Bs[kk * BS + c0 + q * 4] = *(const float4*)(row + q * 4);
    }
    __syncthreads();
#pragma unroll
    for (int k4 = 0; k4 < BK; k4 += 4) {
      const int k0 = k4 + khalf;
      v2f b;
      b.x = Bs[k0 * BS + cw];
      b.y = Bs[(k0 + 1) * BS + cw];
#pragma unroll
      for (int ms = 0; ms < 4; ++ms) {
        v2f a;
        a.x = As[(ms * 16 + lrow) * AS + k0];
        a.y = As[(ms * 16 + lrow) * AS + k0 + 1];
        acc[ms] = wmma4(a, b, acc[ms]);
      }
    }
    __syncthreads();
  }

  const int nglob = nb * BN + cw;
  const int mup = (lane >> 4) << 3;
#pragma unroll
  for (int ms = 0; ms < 4; ++ms) {
#pragma unroll
    for (int i = 0; i < 8; ++i) {
      const int mloc = ms * 16 + i + mup;
      if (mode == 0) {
        const size_t idx = (size_t)rowTok[mloc] * kH + nglob;
        outp[idx] = resid[idx] + acc[ms][i];
      } else if ((mb * BM + mloc) < cnt) {
        const size_t idx = (size_t)rowTok[mloc] * kH + nglob;
        outp[idx] += acc[ms][i];  // single writer per element (top-1 routing)
      }
    }
  }
}

// ---------------------------------------------------------------------------
extern "C" void kernel_launch(void* const* d_in, const int* in_sizes, int n_in,
                              void* d_out, int out_size, void* d_ws,
                              size_t ws_size, hipStream_t stream) {
  (void)in_sizes; (void)n_in; (void)out_size; (void)ws_size;
  const float* x   = (const float*)d_in[0];  // [T,H]
  const float* nw  = (const float*)d_in[1];  // [H]
  const float* rw  = (const float*)d_in[2];  // [E,H]
  const float* gup = (const float*)d_in[3];  // [E,H,2I]
  const float* dp  = (const float*)d_in[4];  // [E,I,H]
  const float* sgw = (const float*)d_in[5];  // [I,H]
  const float* suw = (const float*)d_in[6];  // [I,H]
  const float* sdw = (const float*)d_in[7];  // [H,I]
  float* out = (float*)d_out;

  // Workspace: h (16MB) + act (32MB, reused shared->routed) + routing state.
  float* hbuf  = (float*)d_ws;
  float* act   = hbuf + (size_t)kT * kH;
  float* score = act + (size_t)kT * kI;
  int*   eid   = (int*)(score + kT);
  int*   tlist = eid + kT;
  int*   ecnt  = tlist + kT;
  int*   eoff  = ecnt + kE;

  rmsnorm_router_kernel<<<kT, 256, 0, stream>>>(x, nw, rw, hbuf, score, eid);
  bucket_kernel<<<1, 256, 0, stream>>>(eid, tlist, ecnt, eoff);

  // Shared expert: act = silu(h Wg^T) * (h Wu^T); out = x + act Wd^T
  gateup_kernel<<<dim3(kI / BN, kT / BM, 1), 256, 0, stream>>>(
      hbuf, score, tlist, ecnt, eoff, sgw, suw, nullptr, act, 0);
  down_kernel<<<dim3(kH / BN, kT / BM, 1), 256, 0, stream>>>(
      act, tlist, ecnt, eoff, sdw, nullptr, x, out, 0);

  // Routed experts (top-1, grouped GEMM): act reused, out accumulated.
  gateup_kernel<<<dim3(kI / BN, kT / BM, kE), 256, 0, stream>>>(
      hbuf, score, tlist, ecnt, eoff, nullptr, nullptr, gup, act, 1);
  down_kernel<<<dim3(kH / BN, kT / BM, kE), 256, 0, stream>>>(
      act, tlist, ecnt, eoff, nullptr, dp, nullptr, out, 1);
}